// RegionLoss_64914135712052
// MI455X (gfx1250) — compile-verified
//
#include <hip/hip_runtime.h>

// ---------------- problem constants (from reference) ----------------
#define NCLS   13
#define NANCH  5
#define NHH    76
#define NWW    76
#define NBB    32
#define CHANS  (19 + NCLS)              // 32 channels per anchor
#define PLANE  (NHH * NWW)              // 5776
#define CELLS  (NBB * NANCH * PLANE)    // 924160  (== 3610 * 256 exactly)
#define BLOCK  256
#define NBLK   (CELLS / BLOCK)          // 3610
#define TGT_F4 ((BLOCK * 19) / 4)       // 1216 float4 per block slice

typedef float v2f __attribute__((ext_vector_type(2)));
typedef float v8f __attribute__((ext_vector_type(8)));

// Wave32 all-reduce sum via V_WMMA_F32_16X16X4_F32.
// A(16x4): vgpr0 = per-lane value, vgpr1 = 0  ->  A[m][0]=v(m), A[m][2]=v(m+16)
// B(4x16): all ones.  D[m][n] = v(m) + v(m+16) for every n.
// Per-lane sum of the 8 D vgprs covers one half of m; one shfl_xor(16) finishes.
__device__ __forceinline__ float wave_sum_wmma(float v) {
    v2f a; a.x = v;   a.y = 0.0f;
    v2f b; b.x = 1.0f; b.y = 1.0f;
    v8f c = {};
    v8f d = __builtin_amdgcn_wmma_f32_16x16x4_f32(false, a, false, b,
                                                  (short)0, c, false, false);
    float s = ((d[0] + d[1]) + (d[2] + d[3])) + ((d[4] + d[5]) + (d[6] + d[7]));
    s += __shfl_xor(s, 16, 32);
    return s;
}

__launch_bounds__(BLOCK)
__global__ void region_loss_partial(const float* __restrict__ out,
                                    const float* __restrict__ tgt,
                                    float* __restrict__ partial)
{
    __shared__ float4 tgt_s4[TGT_F4];          // 19456 B staged target slice
    __shared__ float  wsum[BLOCK / 32];

    const int tid   = threadIdx.x;
    const int cell0 = blockIdx.x * BLOCK;

    // ---- cooperative, coalesced b128 stage of this block's target slice ----
    const float4* tsrc = (const float4*)(tgt + (size_t)cell0 * 19);
    #pragma unroll
    for (int r = 0; r < 5; ++r) {
        int i = tid + r * BLOCK;
        if (i < TGT_F4) tgt_s4[i] = tsrc[i];
    }
    __syncthreads();
    const float* t = ((const float*)tgt_s4) + tid * 19;   // stride 19: 0 bank conflicts

    // ---- per-cell addressing ----
    const int cell  = cell0 + tid;
    const int chunk = cell / PLANE;            // b*NANCH + a
    const int rem   = cell - chunk * PLANE;    // h*NWW + w
    const int h     = rem / NWW;
    const int w     = rem - h * NWW;
    const float* op = out + (size_t)chunk * (CHANS * PLANE) + rem;  // channel stride PLANE

    const float t0  = t[0];
    const int   cls = (int)t0;
    const float obj = (t0 != 0.0f) ? 1.0f : 0.0f;

    // ---- class logits: max, logsumexp, select logit[cls] without dyn-index ----
    float l[NCLS];
    float m = -1e30f;
    #pragma unroll
    for (int c = 0; c < NCLS; ++c) { l[c] = op[c * PLANE]; m = fmaxf(m, l[c]); }
    float se = 0.0f, lc = 0.0f;
    #pragma unroll
    for (int c = 0; c < NCLS; ++c) {
        se += __expf(l[c] - m);
        lc  = (c == cls) ? l[c] : lc;
    }
    const float lse  = m + __logf(se);
    const float conf = op[NCLS * PLANE];

    // ---- coord + conf terms ----
    const float gxf = (float)w, gyf = (float)h;
    float lcoord = 0.0f, tconf = 0.0f;
    #pragma unroll
    for (int k = 0; k < 9; ++k) {
        float x = op[(NCLS + 1 + 2 * k) * PLANE];
        float y = op[(NCLS + 2 + 2 * k) * PLANE];
        if (k == 0) {                               // only first pair is sigmoided
            x = 1.0f / (1.0f + __expf(-x));
            y = 1.0f / (1.0f + __expf(-y));
        }
        const float tx = t[1 + 2 * k] - gxf;        // txs
        const float ty = t[2 + 2 * k] - gyf;        // tys
        const float dx = tx - (x + gxf);            // txs - pxs
        const float dy = ty - (y + gyf);            // tys - pys
        const float dt = sqrtf(dx * dx + dy * dy);
        tconf += (dt < 30.0f) ? __expf(2.0f - dt * (1.0f / 15.0f)) : 0.0f;
        const float ex = x - tx, ey = y - ty;
        lcoord += ex * ex + ey * ey;
    }
    tconf *= (1.0f / 9.0f);

    float v  = 0.5f * obj * lcoord;                 // COORD_SCALE = 1
    const float dc = 0.1f * (conf - tconf);         // NOOBJECT_SCALE = 0.1
    v += 0.5f * dc * dc;
    v += obj * (lse - lc);                          // CLASS_SCALE = 1, NLL

    // ---- reductions: wave (WMMA) -> block (LDS) -> global partials ----
    const float s = wave_sum_wmma(v);               // EXEC all-1s here
    const int lane = tid & 31, wave = tid >> 5;
    if (lane == 0) wsum[wave] = s;
    __syncthreads();
    if (tid == 0) {
        float bs = 0.0f;
        #pragma unroll
        for (int i = 0; i < BLOCK / 32; ++i) bs += wsum[i];
        partial[blockIdx.x] = bs;
    }
}

__launch_bounds__(1024)
__global__ void reduce_partials(const float* __restrict__ partial, int n,
                                float* __restrict__ outp)
{
    __shared__ float ws[32];
    float s = 0.0f;
    for (int i = threadIdx.x; i < n; i += 1024) s += partial[i];
    #pragma unroll
    for (int off = 16; off > 0; off >>= 1) s += __shfl_xor(s, off, 32);
    const int lane = threadIdx.x & 31, wave = threadIdx.x >> 5;
    if (lane == 0) ws[wave] = s;
    __syncthreads();
    if (threadIdx.x == 0) {
        float tot = 0.0f;
        #pragma unroll
        for (int i = 0; i < 32; ++i) tot += ws[i];
        outp[0] = tot;
    }
}

extern "C" void kernel_launch(void* const* d_in, const int* in_sizes, int n_in,
                              void* d_out, int out_size, void* d_ws, size_t ws_size,
                              hipStream_t stream)
{
    (void)in_sizes; (void)n_in; (void)out_size; (void)ws_size;
    const float* out_t = (const float*)d_in[0];   // (32, 160, 76, 76) f32
    const float* tgt_t = (const float*)d_in[1];   // (32, 5, 76, 76, 19) f32
    float* partials    = (float*)d_ws;            // NBLK floats of scratch
    float* loss        = (float*)d_out;           // scalar f32

    region_loss_partial<<<NBLK, BLOCK, 0, stream>>>(out_t, tgt_t, partials);
    reduce_partials<<<1, 1024, 0, stream>>>(partials, NBLK, loss);
}